// TaylorSelfAttentionModule_65515431133909
// MI455X (gfx1250) — compile-verified
//
#include <hip/hip_runtime.h>

typedef __attribute__((ext_vector_type(2))) float v2f;
typedef __attribute__((ext_vector_type(8))) float v8f;

#define IN_DIM 256
#define K_TOT  33152          // IN_DIM + TRI, divisible by 4
#define NQ     (K_TOT / 4)    // 8288 K-chunks of 4
#define KSPLIT 4
#define NQ_P   (NQ / KSPLIT)  // 2072
#define N_TOT  8192           // NHEAD * S * D_MODEL
#define BATCH  8
#define SEQ    32
#define OUTF   64

// featsA layout: per K-chunk q (4 K-values x 16 rows = 64 floats), WMMA A-fragment order:
// lane L holds row M = L%16; VGPR0 = K = 4q + (L<16?0:2); VGPR1 = K+1.
// float index = q*64 + L*2 + (k&1), with L = (k&2 ? 16 : 0) + m.
__device__ __forceinline__ int fa_idx(int m, int k) {
    int q = k >> 2;
    int lane = ((k & 2) ? 16 : 0) + m;
    return q * 64 + lane * 2 + (k & 1);
}

// grid: 257 blocks x 256 threads. Block i<256: triu row i (pairs (i,j), j>=i).
// Block 256: linear features k=0..255. Rows m=8..15 zero-padded (M pad to 16).
__global__ void build_featsA(const float* __restrict__ x, float* __restrict__ featsA) {
    int i = blockIdx.x;
    int j = threadIdx.x;
    if (i == IN_DIM) {
        int k = j;
        #pragma unroll
        for (int m = 0; m < 16; ++m) {
            float v = (m < BATCH) ? x[m * IN_DIM + k] : 0.0f;
            featsA[fa_idx(m, k)] = v;
        }
    } else if (j >= i) {
        int r_i = i * IN_DIM - (i * (i - 1)) / 2;   // triu row start (row-major triu)
        int k = IN_DIM + r_i + (j - i);
        #pragma unroll
        for (int m = 0; m < 16; ++m) {
            float v = (m < BATCH) ? x[m * IN_DIM + i] * x[m * IN_DIM + j] : 0.0f;
            featsA[fa_idx(m, k)] = v;
        }
    }
}

// One wave (32 threads) per (16-row N-tile, K-split part).
// grid = (N_TOT/16, KSPLIT). head_part layout: part p -> head + p*16*N_TOT, [M=16][N_TOT].
__global__ void __launch_bounds__(32) head_gemm(const float* __restrict__ featsA,
                                                const float* __restrict__ W,     // [N_TOT][K_TOT]
                                                float* __restrict__ head) {
    const int lane = threadIdx.x;
    const int m16  = lane & 15;
    const int hi   = lane >> 4;        // 0: K={0,1}, 1: K={2,3} within a chunk
    const int n0   = blockIdx.x * 16;
    const int part = blockIdx.y;

    const float* wrow = W + (size_t)(n0 + m16) * K_TOT + hi * 2;  // B: lane = column N
    const float* ap   = featsA + lane * 2;                         // A: lane = row M

    v8f c = {};  // D/C 16x16 f32 accumulator (8 VGPRs)
    const int q0 = part * NQ_P;
    const int q1 = q0 + NQ_P;
    #pragma unroll 4
    for (int q = q0; q < q1; ++q) {
        v2f a = *(const v2f*)(ap   + q * 64);  // A[16x4] fragment, coalesced b64
        v2f b = *(const v2f*)(wrow + q * 4);   // B[4x16] fragment, 16B/row/step
        c = __builtin_amdgcn_wmma_f32_16x16x4_f32(
                /*neg_a=*/false, a, /*neg_b=*/false, b,
                /*c_mod=*/(short)0, c, /*reuse_a=*/false, /*reuse_b=*/false);
    }

    // C/D layout: VGPR g holds element (M = g + 8*hi, N = m16)
    float* hp = head + (size_t)part * 16 * N_TOT;
    #pragma unroll
    for (int g = 0; g < 8; ++g) {
        int m = g + hi * 8;
        hp[(size_t)m * N_TOT + n0 + m16] = c[g];
    }
}

// grid = B*S blocks x OUTF threads. out[b,s,of] = b_out[of] + sum_c concat[c]*W_out[of,c]
__global__ void out_proj(const float* __restrict__ head,     // KSPLIT partials, [16][N_TOT] each
                         const float* __restrict__ b_heads,  // [N_TOT]
                         const float* __restrict__ W_out,    // [OUTF][256]
                         const float* __restrict__ b_out,    // [OUTF]
                         float* __restrict__ out) {
    int bs = blockIdx.x;
    int b = bs >> 5, s = bs & 31;
    int of = threadIdx.x;
    const float* wr = W_out + of * 256;
    float acc = b_out[of];
    #pragma unroll 4
    for (int cidx = 0; cidx < 256; ++cidx) {
        int h = cidx >> 6, d = cidx & 63;
        int n = h * 2048 + s * 64 + d;          // head_out flat index for this (h, s, d)
        float hv = 0.0f;
        #pragma unroll
        for (int p = 0; p < KSPLIT; ++p)
            hv += head[(size_t)p * 16 * N_TOT + (size_t)b * N_TOT + n];
        acc += (hv + b_heads[n]) * wr[cidx];
    }
    out[bs * OUTF + of] = acc;
}

extern "C" void kernel_launch(void* const* d_in, const int* in_sizes, int n_in,
                              void* d_out, int out_size, void* d_ws, size_t ws_size,
                              hipStream_t stream) {
    const float* input_  = (const float*)d_in[0];  // (8,32,8)
    const float* W_heads = (const float*)d_in[1];  // (4,2048,33152)
    const float* b_heads = (const float*)d_in[2];  // (4,2048)
    const float* W_out   = (const float*)d_in[3];  // (64,256)
    const float* b_out   = (const float*)d_in[4];  // (64,)
    float* out = (float*)d_out;

    float* featsA = (float*)d_ws;                 // 16*K_TOT floats = 2.12 MB
    float* head   = featsA + 16 * K_TOT;          // KSPLIT * 16 * N_TOT floats = 2.0 MB

    build_featsA<<<IN_DIM + 1, IN_DIM, 0, stream>>>(input_, featsA);

    dim3 g(N_TOT / 16, KSPLIT);
    head_gemm<<<g, 32, 0, stream>>>(featsA, W_heads, head);

    out_proj<<<BATCH * SEQ, OUTF, 0, stream>>>(head, b_heads, W_out, b_out, out);
}